// SpatialEncoder_71399536329196
// MI455X (gfx1250) — compile-verified
//
#include <hip/hip_runtime.h>

typedef __attribute__((ext_vector_type(2))) float v2f;
typedef __attribute__((ext_vector_type(8))) float v8f;

#define BM 64
#define BN 64
#define BKK 16
#define GEMM_THREADS 128

enum { OPB_N = 0, OPB_T = 1, OPB_CONV3 = 2 };
enum { EPI_BIAS = 0, EPI_GAMMA_RES = 1 };

// -----------------------------------------------------------------------------
// Tiled fp32 WMMA GEMM:  C[b][m][n] = sum_k A[b][m][k] * B[b][k][n]  (+epilogue)
//   TRANSA=1: logical A[m][k] = Aptr[k*lda + m]
//   OPB_N:    B[k][n] = Bptr[k*ldb + n]
//   OPB_T:    B[k][n] = Bptr[n*ldb + k]
//   OPB_CONV3: implicit im2col of 3x3 pad=1 conv; Bptr = X[b] as [Cin][H*W];
//              k = ci*9 + (dy+1)*3 + (dx+1); n = y*W + x
// 4 waves / block; each wave computes 2x2 16x16 tiles with
// V_WMMA_F32_16X16X4_F32 (fp32 in / fp32 acc -> matches fp32 reference).
// Register-staged LDS double buffering, one barrier per K-chunk.
// -----------------------------------------------------------------------------
template <int TRANSA, int OPB, int EPI>
__global__ __launch_bounds__(GEMM_THREADS)
void gemm_f32_wmma(const float* __restrict__ A, int lda, long long sAb,
                   const float* __restrict__ Bp, int ldb, long long sBb,
                   float* __restrict__ C, int ldc, long long sCb,
                   int M, int N, int K,
                   const float* __restrict__ bias,
                   const float* __restrict__ gamma,
                   const float* __restrict__ res, long long sRb,
                   int convH, int convW)
{
    __shared__ float As[2][BM * BKK];   // As[m][k]
    __shared__ float Bs[2][BN * BKK];   // Bs[n][k] (B tile stored transposed)

    const int b = blockIdx.z;
    const float* Ab = A + (long long)b * sAb;
    const float* Bb = Bp + (long long)b * sBb;
    float* Cb = C + (long long)b * sCb;

    const int tileM = blockIdx.y * BM;
    const int tileN = blockIdx.x * BN;
    const int t    = threadIdx.x;
    const int lane = t & 31;
    const int wave = t >> 5;
    const int wm   = wave >> 1;   // 0..1
    const int wn   = wave & 1;    // 0..1

    // ---- per-thread loader constants (hoisted out of the K loop) ----
    // A, TRANSA=0: row am = t>>1, cols ak0..ak0+7 contiguous
    const int amN  = t >> 1;
    const int akN  = (t & 1) * 8;
    // A, TRANSA=1 / B normal|conv: k-row = t>>3, 8 contiguous m/n
    const int krow = t >> 3;               // 0..15
    const int c0   = (t * 8) & 63;         // 0,8,...,56
    // B, OPB_T: row n = t>>1, cols k contiguous
    const int bnT  = t >> 1;
    const int bkT  = (t & 1) * 8;

    float aReg[8], bReg[8];

    auto loadA = [&](int k0) {
        if (TRANSA == 0) {
            const int gm = tileM + amN;
            const bool mok = gm < M;
            const float* p = Ab + (long long)gm * lda + (k0 + akN);
            #pragma unroll
            for (int i = 0; i < 8; ++i)
                aReg[i] = (mok && (k0 + akN + i) < K) ? p[i] : 0.f;
        } else {
            const int gk = k0 + krow;
            const bool kok = gk < K;
            const float* p = Ab + (long long)gk * lda + (tileM + c0);
            #pragma unroll
            for (int i = 0; i < 8; ++i)
                aReg[i] = (kok && (tileM + c0 + i) < M) ? p[i] : 0.f;
        }
    };

    auto loadB = [&](int k0) {
        if (OPB == OPB_N) {
            const int gk = k0 + krow;
            const bool kok = gk < K;
            const float* p = Bb + (long long)gk * ldb + (tileN + c0);
            #pragma unroll
            for (int i = 0; i < 8; ++i)
                bReg[i] = (kok && (tileN + c0 + i) < N) ? p[i] : 0.f;
        } else if (OPB == OPB_T) {
            const int gn = tileN + bnT;
            const bool nok = gn < N;
            const float* p = Bb + (long long)gn * ldb + (k0 + bkT);
            #pragma unroll
            for (int i = 0; i < 8; ++i)
                bReg[i] = (nok && (k0 + bkT + i) < K) ? p[i] : 0.f;
        } else { // OPB_CONV3: k is per-thread constant -> tap math once
            const int gk  = k0 + krow;
            const bool kok = gk < K;
            const int ci  = gk / 9;
            const int tap = gk - ci * 9;
            const int dy  = tap / 3 - 1;
            const int dx  = tap - (tap / 3) * 3 - 1;
            const float* p = Bb + (long long)ci * (convH * convW);
            #pragma unroll
            for (int i = 0; i < 8; ++i) {
                const int n = tileN + c0 + i;
                const int y = n / convW;
                const int x = n - y * convW;
                const int yy = y + dy, xx = x + dx;
                const bool ok = kok && n < N &&
                                (unsigned)yy < (unsigned)convH &&
                                (unsigned)xx < (unsigned)convW;
                bReg[i] = ok ? p[yy * convW + xx] : 0.f;
            }
        }
    };

    auto storeTile = [&](int buf) {
        if (TRANSA == 0) {
            #pragma unroll
            for (int i = 0; i < 8; ++i) As[buf][amN * BKK + akN + i] = aReg[i];
        } else {
            #pragma unroll
            for (int i = 0; i < 8; ++i) As[buf][(c0 + i) * BKK + krow] = aReg[i];
        }
        if (OPB == OPB_T) {
            #pragma unroll
            for (int i = 0; i < 8; ++i) Bs[buf][bnT * BKK + bkT + i] = bReg[i];
        } else {
            #pragma unroll
            for (int i = 0; i < 8; ++i) Bs[buf][(c0 + i) * BKK + krow] = bReg[i];
        }
    };

    v8f acc[2][2] = {};

    loadA(0);
    loadB(0);
    storeTile(0);

    int cur = 0;
    for (int k0 = 0; k0 < K; k0 += BKK) {
        __syncthreads();
        const bool more = (k0 + BKK) < K;
        if (more) { loadA(k0 + BKK); loadB(k0 + BKK); }

        // ---- 4 K-steps of V_WMMA_F32_16X16X4_F32 on the current LDS tile ----
        #pragma unroll
        for (int ks = 0; ks < 4; ++ks) {
            const int klocal = 4 * ks + 2 * (lane >> 4);
            v2f af[2], bf[2];
            #pragma unroll
            for (int mi = 0; mi < 2; ++mi) {
                const int mlocal = wm * 32 + mi * 16 + (lane & 15);
                af[mi] = *reinterpret_cast<const v2f*>(&As[cur][mlocal * BKK + klocal]);
            }
            #pragma unroll
            for (int ni = 0; ni < 2; ++ni) {
                const int nlocal = wn * 32 + ni * 16 + (lane & 15);
                bf[ni] = *reinterpret_cast<const v2f*>(&Bs[cur][nlocal * BKK + klocal]);
            }
            #pragma unroll
            for (int mi = 0; mi < 2; ++mi)
                #pragma unroll
                for (int ni = 0; ni < 2; ++ni)
                    acc[mi][ni] = __builtin_amdgcn_wmma_f32_16x16x4_f32(
                        false, af[mi], false, bf[ni],
                        (short)0, acc[mi][ni], false, false);
        }

        if (more) storeTile(cur ^ 1);
        cur ^= 1;
    }

    // ---- epilogue ----
    float g = 0.f;
    const float* resb = nullptr;
    if (EPI == EPI_GAMMA_RES) {
        g = gamma[0];
        resb = res + (long long)b * sRb;
    }
    #pragma unroll
    for (int mi = 0; mi < 2; ++mi)
        #pragma unroll
        for (int ni = 0; ni < 2; ++ni) {
            const int nIdx  = tileN + wn * 32 + ni * 16 + (lane & 15);
            const int mBase = tileM + wm * 32 + mi * 16 + 8 * (lane >> 4);
            #pragma unroll
            for (int r = 0; r < 8; ++r) {
                const int mIdx = mBase + r;
                if (mIdx < M && nIdx < N) {
                    float v = acc[mi][ni][r];
                    if (EPI == EPI_BIAS) {
                        if (bias) v += bias[mIdx];
                    } else {
                        v = g * v + resb[(long long)mIdx * ldc + nIdx];
                    }
                    Cb[(long long)mIdx * ldc + nIdx] = v;
                }
            }
        }
}

// -----------------------------------------------------------------------------
// In-place row softmax of (max(E)-E) == softmax(-E), stable form:
//   att[m] = exp(min(E) - E[m]) / sum_m exp(min(E) - E[m])
// One 256-thread block per row (row length n = 1600 -> 7 regs/thread).
// -----------------------------------------------------------------------------
__global__ __launch_bounds__(256)
void softmax_negmin_inplace(float* __restrict__ E, int n)
{
    constexpr int VPT = 7;
    const long long row = blockIdx.x;
    float* e = E + row * (long long)n;
    const int t = threadIdx.x;

    float vals[VPT];
    float lmin = 3.4e38f;
    #pragma unroll
    for (int ii = 0; ii < VPT; ++ii) {
        const int i = t + ii * 256;
        const float v = (i < n) ? e[i] : 3.4e38f;
        vals[ii] = v;
        lmin = fminf(lmin, v);
    }
    __shared__ float red[256];
    red[t] = lmin; __syncthreads();
    for (int s = 128; s > 0; s >>= 1) {
        if (t < s) red[t] = fminf(red[t], red[t + s]);
        __syncthreads();
    }
    const float mn = red[0];
    __syncthreads();

    float lsum = 0.f;
    #pragma unroll
    for (int ii = 0; ii < VPT; ++ii) {
        const int i = t + ii * 256;
        const float ex = (i < n) ? __expf(mn - vals[ii]) : 0.f;
        vals[ii] = ex;
        lsum += ex;
    }
    red[t] = lsum; __syncthreads();
    for (int s = 128; s > 0; s >>= 1) {
        if (t < s) red[t] += red[t + s];
        __syncthreads();
    }
    const float inv = 1.f / red[0];
    #pragma unroll
    for (int ii = 0; ii < VPT; ++ii) {
        const int i = t + ii * 256;
        if (i < n) e[i] = vals[ii] * inv;
    }
}

// -----------------------------------------------------------------------------
extern "C" void kernel_launch(void* const* d_in, const int* in_sizes, int n_in,
                              void* d_out, int out_size, void* d_ws, size_t ws_size,
                              hipStream_t stream) {
    (void)in_sizes; (void)n_in; (void)out_size; (void)ws_size;

    const float* x1         = (const float*)d_in[0];   // [8,425,40,40]
    const float* x2         = (const float*)d_in[1];   // [8,2048,40,40]
    const float* ssam_w1    = (const float*)d_in[2];   // [425,425,1,1]
    const float* ssam_b1    = (const float*)d_in[3];
    const float* ssam_w2    = (const float*)d_in[4];   // [425,425,3,3]
    const float* ssam_b2    = (const float*)d_in[5];
    const float* ssam_gamma = (const float*)d_in[6];
    const float* scam_w     = (const float*)d_in[7];   // [425,2048,1,1]
    const float* scam_b     = (const float*)d_in[8];
    const float* scam_gamma = (const float*)d_in[9];
    const float* out_w1     = (const float*)d_in[10];  // [425,850,1,1]
    const float* out_b1     = (const float*)d_in[11];
    const float* out_w2     = (const float*)d_in[12];  // [425,425,3,3]
    const float* out_b2     = (const float*)d_in[13];
    const float* out_w3     = (const float*)d_in[14];  // [425,425,1,1]
    const float* out_b3     = (const float*)d_in[15];
    float* out = (float*)d_out;

    const int Bb = 8, C1 = 425, C2 = 2048, H = 40, W = 40, NN = 1600;
    const long long sCh  = (long long)C1 * NN;        // per-batch [425,1600]
    const long long sE   = (long long)NN * NN;        // per-batch [1600,1600]
    const long long sCat = (long long)(2 * C1) * NN;  // per-batch [850,1600]

    float* buf0   = (float*)d_ws;               // [8,425,1600]  21.8 MB
    float* buf1   = buf0 + (long long)Bb * sCh; // [8,425,1600]  21.8 MB
    float* bufE   = buf1 + (long long)Bb * sCh; // [8,1600,1600] 81.9 MB
    float* bufCat = bufE + (long long)Bb * sE;  // [8,850,1600]  43.5 MB

    const dim3 blk(GEMM_THREADS);
    auto grid = [](int M, int N, int B) {
        return dim3((N + BN - 1) / BN, (M + BM - 1) / BM, B);
    };

    // 1) h1 = conv1x1(x1, ssam_w1) + b1                          -> buf0
    gemm_f32_wmma<0, OPB_N, EPI_BIAS><<<grid(C1, NN, Bb), blk, 0, stream>>>(
        ssam_w1, C1, 0,  x1, NN, sCh,
        buf0, NN, sCh,  C1, NN, C1,
        ssam_b1, nullptr, nullptr, 0, H, W);

    // 2) h2 = conv3x3(h1, ssam_w2) + b2  (implicit im2col)       -> buf1
    gemm_f32_wmma<0, OPB_CONV3, EPI_BIAS><<<grid(C1, NN, Bb), blk, 0, stream>>>(
        ssam_w2, 9 * C1, 0,  buf0, NN, sCh,
        buf1, NN, sCh,  C1, NN, 9 * C1,
        ssam_b2, nullptr, nullptr, 0, H, W);

    // 3) E = h2^T . h2   (energy[n][m])                          -> bufE
    gemm_f32_wmma<1, OPB_N, EPI_BIAS><<<grid(NN, NN, Bb), blk, 0, stream>>>(
        buf1, NN, sCh,  buf1, NN, sCh,
        bufE, NN, sE,  NN, NN, C1,
        nullptr, nullptr, nullptr, 0, H, W);

    // 4) att = softmax(-E) rowwise, in place
    softmax_negmin_inplace<<<Bb * NN, 256, 0, stream>>>(bufE, NN);

    // 5) cat[:, :425] = gamma * (h2 . att^T) + h2                -> bufCat lo
    gemm_f32_wmma<0, OPB_T, EPI_GAMMA_RES><<<grid(C1, NN, Bb), blk, 0, stream>>>(
        buf1, NN, sCh,  bufE, NN, sE,
        bufCat, NN, sCat,  C1, NN, NN,
        nullptr, ssam_gamma, buf1, sCh, H, W);

    // 6) x2p = conv1x1(x2, scam_w) + b                           -> buf0
    gemm_f32_wmma<0, OPB_N, EPI_BIAS><<<grid(C1, NN, Bb), blk, 0, stream>>>(
        scam_w, C2, 0,  x2, NN, (long long)C2 * NN,
        buf0, NN, sCh,  C1, NN, C2,
        scam_b, nullptr, nullptr, 0, H, W);

    // 7) E = x1^T . x2p                                          -> bufE
    gemm_f32_wmma<1, OPB_N, EPI_BIAS><<<grid(NN, NN, Bb), blk, 0, stream>>>(
        x1, NN, sCh,  buf0, NN, sCh,
        bufE, NN, sE,  NN, NN, C1,
        nullptr, nullptr, nullptr, 0, H, W);

    // 8) att = softmax(-E)
    softmax_negmin_inplace<<<Bb * NN, 256, 0, stream>>>(bufE, NN);

    // 9) cat[:, 425:] = gamma * (x2p . att^T) + x1               -> bufCat hi
    gemm_f32_wmma<0, OPB_T, EPI_GAMMA_RES><<<grid(C1, NN, Bb), blk, 0, stream>>>(
        buf0, NN, sCh,  bufE, NN, sE,
        bufCat + (long long)C1 * NN, NN, sCat,  C1, NN, NN,
        nullptr, scam_gamma, x1, sCh, H, W);

    // 10) o1 = conv1x1(cat, out_w1) + b1                         -> buf1
    gemm_f32_wmma<0, OPB_N, EPI_BIAS><<<grid(C1, NN, Bb), blk, 0, stream>>>(
        out_w1, 2 * C1, 0,  bufCat, NN, sCat,
        buf1, NN, sCh,  C1, NN, 2 * C1,
        out_b1, nullptr, nullptr, 0, H, W);

    // 11) o2 = conv3x3(o1, out_w2) + b2                          -> buf0
    gemm_f32_wmma<0, OPB_CONV3, EPI_BIAS><<<grid(C1, NN, Bb), blk, 0, stream>>>(
        out_w2, 9 * C1, 0,  buf1, NN, sCh,
        buf0, NN, sCh,  C1, NN, 9 * C1,
        out_b2, nullptr, nullptr, 0, H, W);

    // 12) out = conv1x1(o2, out_w3) + b3                         -> d_out
    gemm_f32_wmma<0, OPB_N, EPI_BIAS><<<grid(C1, NN, Bb), blk, 0, stream>>>(
        out_w3, C1, 0,  buf0, NN, sCh,
        out, NN, sCh,  C1, NN, C1,
        out_b3, nullptr, nullptr, 0, H, W);
}